// shotLevelExtractionV2_79431125172928
// MI455X (gfx1250) — compile-verified
//
#include <hip/hip_runtime.h>
#include <hip/hip_bf16.h>

typedef __attribute__((ext_vector_type(2))) float v2f;
typedef __attribute__((ext_vector_type(4))) float v4f;
typedef __attribute__((ext_vector_type(8))) float v8f;

#define SEQ_LEN 32768
#define EMB     1024
#define SHOT    20
#define MPAD    32                   // padded M for WMMA tiles (rows 20..31 zero)
#define SPLIT   1638                 // SEQ_LEN / SHOT
#define SC      8096                 // shot_channel
#define FLATL   (SPLIT * EMB)        // 1677312
#define LC1     2024
#define LC2     2023
#define LC3     2023
#define LC4     2024
#define CATL    (LC1 + LC2 + LC3 + LC4)  // 8094
#define KDIM    SC                   // GEMM K = 8096
#define NDIM    EMB                  // GEMM N = 1024
#define KCHUNKS 8
#define KPER    (KDIM / KCHUNKS)     // 1012 (multiple of 4)

// ---------------------------------------------------------------------------
// Kernel 1: shotAvgPool. One wave32 per output element (s, j).
// Each lane strides the ~207-element adaptive window; wave-reduce via shuffles.
// seq is read with default (temporal) policy so it stays L2-resident for the
// final merge pass (134 MB < 192 MB L2).
// ---------------------------------------------------------------------------
__global__ void __launch_bounds__(256)
shot_pool_kernel(const float* __restrict__ seq, float* __restrict__ seq_shot) {
  int wave = (blockIdx.x * blockDim.x + threadIdx.x) >> 5;
  int lane = threadIdx.x & 31;
  if (wave >= SHOT * SC) return;
  int s = wave / SC;
  int j = wave - s * SC;
  int start = (int)(((long long)j * FLATL) / SC);
  int end   = (int)((((long long)(j + 1) * FLATL) + SC - 1) / SC);
  float sum = 0.f;
  for (int f = start + lane; f < end; f += 32) {
    int k = f >> 10;          // EMB == 1024
    int e = f & 1023;
    // faithful to reference index map (prev-shot wrap, python modulo)
    int t = (k == 0) ? (s * SPLIT)
                     : (((s - 1) * SPLIT + k + SEQ_LEN) % SEQ_LEN);
    sum += seq[t * EMB + e];
  }
  for (int off = 16; off > 0; off >>= 1) sum += __shfl_xor(sum, off, 32);
  if (lane == 0) seq_shot[s * SC + j] = sum / (float)(end - start);
}

// ---------------------------------------------------------------------------
// Kernel 2: dilated convs (stride 4) + concat + adaptive pool.
// Writes a 32-row padded pooled[MPAD][8096]; rows >= 20 are zeros so the WMMA
// kernel needs no per-iteration masking on its second M tile.
// ---------------------------------------------------------------------------
__global__ void __launch_bounds__(256)
conv_pool_kernel(const float* __restrict__ ss,
                 const float* __restrict__ w1, const float* __restrict__ b1,
                 const float* __restrict__ w2, const float* __restrict__ b2,
                 const float* __restrict__ w3, const float* __restrict__ b3,
                 const float* __restrict__ w4, const float* __restrict__ b4,
                 float* __restrict__ pooled) {
  int idx = blockIdx.x * blockDim.x + threadIdx.x;
  if (idx >= MPAD * SC) return;
  int s = idx / SC;
  int j = idx - s * SC;
  if (s >= SHOT) { pooled[idx] = 0.f; return; }   // zero padding rows 20..31
  int start = (int)(((long long)j * CATL) / SC);
  int end   = (int)((((long long)(j + 1) * CATL) + SC - 1) / SC);
  float acc = 0.f;
  for (int t = start; t < end; ++t) {
    float v;
    if (t < LC1) {                         // conv1: dil 1, K 3
      int p = t * 4;
      v = b1[s];
      for (int ci = 0; ci < SHOT; ++ci) {
        const float* wr = w1 + (s * SHOT + ci) * 3;
        const float* xr = ss + ci * SC + p;
        v += wr[0] * xr[0] + wr[1] * xr[1] + wr[2] * xr[2];
      }
    } else if (t < LC1 + LC2) {            // conv2: dil 2, K 3
      int p = (t - LC1) * 4;
      v = b2[s];
      for (int ci = 0; ci < SHOT; ++ci) {
        const float* wr = w2 + (s * SHOT + ci) * 3;
        const float* xr = ss + ci * SC + p;
        v += wr[0] * xr[0] + wr[1] * xr[2] + wr[2] * xr[4];
      }
    } else if (t < LC1 + LC2 + LC3) {      // conv3: dil 3, K 3
      int p = (t - LC1 - LC2) * 4;
      v = b3[s];
      for (int ci = 0; ci < SHOT; ++ci) {
        const float* wr = w3 + (s * SHOT + ci) * 3;
        const float* xr = ss + ci * SC + p;
        v += wr[0] * xr[0] + wr[1] * xr[3] + wr[2] * xr[6];
      }
    } else {                               // conv4: K 1
      int p = (t - LC1 - LC2 - LC3) * 4;
      v = b4[s];
      for (int ci = 0; ci < SHOT; ++ci)
        v += w4[s * SHOT + ci] * ss[ci * SC + p];
    }
    acc += v;
  }
  pooled[idx] = acc / (float)(end - start);
}

// ---------------------------------------------------------------------------
// Kernel 3: FP32 WMMA GEMM partials. out[m][n] = sum_k pooled[m][k]*fcw[n][k].
// One wave per (16-col N-tile, K-chunk); two 16x16 accumulators cover the
// zero-padded M=0..31. Inner loop: 3x global_load_b64 + 2x
// V_WMMA_F32_16X16X4_F32 per K=4 step — no VALU masking in the loop.
// Deterministic: partials written to disjoint slots, reduced in kernel 4.
// ---------------------------------------------------------------------------
__global__ void __launch_bounds__(32)
gemm_wmma_kernel(const float* __restrict__ pooled,
                 const float* __restrict__ fcw,
                 float* __restrict__ part) {
  int lane = threadIdx.x & 31;
  int half = lane >> 4;       // 0: K=0,1  1: K=2,3 within a 4-wide step
  int mr   = lane & 15;       // A row / B column within the 16-tile
  int nbase = blockIdx.x * 16;
  int kc    = blockIdx.y;
  int k0    = kc * KPER;

  v8f acc0 = {};              // M = 0..15
  v8f acc1 = {};              // M = 16..31 (rows >= 20 are zero in pooled)
  const float* arow0 = pooled + (size_t)mr * KDIM;
  const float* arow1 = pooled + (size_t)(mr + 16) * KDIM;
  const float* brow  = fcw + (size_t)(nbase + mr) * KDIM;   // fc_w[n][k], K contiguous

  for (int k = k0; k < k0 + KPER; k += 4) {
    int kk = k + 2 * half;                           // even -> 8B aligned
    v2f a0 = *(const v2f*)(arow0 + kk);
    v2f a1 = *(const v2f*)(arow1 + kk);
    v2f b  = *(const v2f*)(brow + kk);
    acc0 = __builtin_amdgcn_wmma_f32_16x16x4_f32(false, a0, false, b,
                                                 (short)0, acc0, false, false);
    acc1 = __builtin_amdgcn_wmma_f32_16x16x4_f32(false, a1, false, b,
                                                 (short)0, acc1, false, false);
  }

  // C/D layout: VGPR r -> M=r (lanes 0-15) / M=r+8 (lanes 16-31), N = lane%16
  float* pbase = part + (size_t)kc * MPAD * NDIM + nbase + mr;
  for (int r = 0; r < 8; ++r) {
    int row = r + 8 * half;
    pbase[(size_t)row * NDIM]        = acc0[r];
    pbase[(size_t)(row + 16) * NDIM] = acc1[r];
  }
}

// ---------------------------------------------------------------------------
// Kernel 4: reduce K-chunk partials + bias -> tanh -> LayerNorm. Block per row.
// ---------------------------------------------------------------------------
__global__ void __launch_bounds__(256)
fc_tanh_ln_kernel(const float* __restrict__ part, const float* __restrict__ fcb,
                  const float* __restrict__ lng, const float* __restrict__ lnb,
                  float* __restrict__ h) {
  int s = blockIdx.x;
  int tid = threadIdx.x;
  __shared__ float redA[256];
  __shared__ float redB[256];
  float t[4];
  float sum = 0.f, sq = 0.f;
  for (int i = 0; i < 4; ++i) {
    int c = tid + i * 256;
    float v = fcb[c];
    for (int kc = 0; kc < KCHUNKS; ++kc)
      v += part[(size_t)(kc * MPAD + s) * NDIM + c];
    float tv = tanhf(v);
    t[i] = tv;
    sum += tv; sq += tv * tv;
  }
  redA[tid] = sum; redB[tid] = sq;
  __syncthreads();
  for (int off = 128; off > 0; off >>= 1) {
    if (tid < off) { redA[tid] += redA[tid + off]; redB[tid] += redB[tid + off]; }
    __syncthreads();
  }
  float mu  = redA[0] * (1.f / NDIM);
  float var = redB[0] * (1.f / NDIM) - mu * mu;
  float inv = rsqrtf(var + 1e-5f);
  for (int i = 0; i < 4; ++i) {
    int c = tid + i * 256;
    h[(size_t)s * NDIM + c] = (t[i] - mu) * inv * lng[c] + lnb[c];
  }
}

// ---------------------------------------------------------------------------
// Kernel 5: mergePooling — out = seq + h[min(t/1638,19)], 128-bit vectorized.
// Dominant memory pass: 134 MB read + 134 MB write. Both streams are
// non-temporal (last use of seq, write-once out) to preserve L2 for the rest.
// Uses native ext-vector v4f so the nontemporal builtins accept the pointers.
// ---------------------------------------------------------------------------
__global__ void __launch_bounds__(256)
merge_kernel(const float* __restrict__ seq, const float* __restrict__ h,
             float* __restrict__ out) {
  int vi = blockIdx.x * blockDim.x + threadIdx.x;   // v4f index
  int t  = vi >> 8;                                 // EMB/4 == 256 vec4 per row
  int e4 = vi & 255;
  int row = t / SPLIT; if (row > SHOT - 1) row = SHOT - 1;
  v4f a = __builtin_nontemporal_load(((const v4f*)seq) + vi);
  v4f b = ((const v4f*)(h + (size_t)row * EMB))[e4];  // hot, keep cached
  v4f o = a + b;
  __builtin_nontemporal_store(o, ((v4f*)out) + vi);
}

// ---------------------------------------------------------------------------
extern "C" void kernel_launch(void* const* d_in, const int* in_sizes, int n_in,
                              void* d_out, int out_size, void* d_ws, size_t ws_size,
                              hipStream_t stream) {
  const float* seq  = (const float*)d_in[0];
  const float* w1   = (const float*)d_in[1];
  const float* b1   = (const float*)d_in[2];
  const float* w2   = (const float*)d_in[3];
  const float* b2   = (const float*)d_in[4];
  const float* w3   = (const float*)d_in[5];
  const float* b3   = (const float*)d_in[6];
  const float* w4   = (const float*)d_in[7];
  const float* b4   = (const float*)d_in[8];
  const float* fcw  = (const float*)d_in[9];
  const float* fcb  = (const float*)d_in[10];
  const float* lng  = (const float*)d_in[11];
  const float* lnb  = (const float*)d_in[12];
  float* out = (float*)d_out;

  char* ws = (char*)d_ws;
  float* seq_shot = (float*)(ws);                 // 20*8096 f32   = 647680 B
  float* pooled   = (float*)(ws + 647680);        // 32*8096 f32   = 1036288 B
  float* part     = (float*)(ws + 1683968);       // 8*32*1024 f32 = 1048576 B
  float* h        = (float*)(ws + 2732544);       // 20*1024 f32

  // 1) shot pooling: 161920 outputs, 8 waves (256 thr) per block
  shot_pool_kernel<<<(SHOT * SC) / 8, 256, 0, stream>>>(seq, seq_shot);

  // 2) convs + concat + adaptive pool (writes zero-padded rows 20..31)
  conv_pool_kernel<<<(MPAD * SC) / 256, 256, 0, stream>>>(
      seq_shot, w1, b1, w2, b2, w3, b3, w4, b4, pooled);

  // 3) FP32 WMMA GEMM partials: 64 N-tiles x 8 K-chunks, one wave each
  gemm_wmma_kernel<<<dim3(NDIM / 16, KCHUNKS), 32, 0, stream>>>(pooled, fcw, part);

  // 4) reduce partials + bias + tanh + LayerNorm
  fc_tanh_ln_kernel<<<SHOT, 256, 0, stream>>>(part, fcb, lng, lnb, h);

  // 5) broadcast add over full sequence (non-temporal streaming pass)
  merge_kernel<<<(SEQ_LEN * EMB / 4) / 256, 256, 0, stream>>>(seq, h, out);
}